// CircuitGraphConv_41678362640893
// MI455X (gfx1250) — compile-verified
//
#include <hip/hip_runtime.h>
#include <hip/hip_bf16.h>

typedef __attribute__((ext_vector_type(2))) float v2f;
typedef __attribute__((ext_vector_type(8))) float v8f;
typedef __attribute__((ext_vector_type(4))) int v4i;

#define N_NODES    50000
#define N_EDGES    800000
#define IN_FEAT    128
#define INTER_DIM  128
#define OUT_FEAT   128
#define K1         131   // IN_FEAT + EDGE_W_DIM
#define K2         256   // IN_FEAT + INTER_DIM
#define EDGE_TILES (N_EDGES / 16)
#define EDGE_BLOCKS 6250 // 8 tiles per persistent block

// ---- CDNA5 async global->LDS staging (guarded; fallback = reg pipeline) ----
#if defined(__HIP_DEVICE_COMPILE__) && defined(__gfx1250__) && \
    __has_builtin(__builtin_amdgcn_global_load_async_to_lds_b128)
#define HAVE_ASYNC_LDS 1
#else
#define HAVE_ASYNC_LDS 0
#endif

#if HAVE_ASYNC_LDS
// probed signature: (int4 addrspace(1)*, int4 addrspace(3)*, imm offset, imm cpol)
typedef __attribute__((address_space(1))) v4i gas_v4i;
typedef __attribute__((address_space(3))) v4i las_v4i;
__device__ __forceinline__ void async_b128(const float* g, float* l) {
  __builtin_amdgcn_global_load_async_to_lds_b128((gas_v4i*)g, (las_v4i*)l, 0, 0);
}
__device__ __forceinline__ void wait_async0() {
#if __has_builtin(__builtin_amdgcn_s_wait_asynccnt)
  __builtin_amdgcn_s_wait_asynccnt(0);
#else
  asm volatile("s_wait_asynccnt 0x0" ::: "memory");
#endif
}
#endif

// fp32 WMMA: D(16x16,f32) = A(16x4,f32) * B(4x16,f32) + C
__device__ __forceinline__ v8f wmma4(v2f a, v2f b, v8f c) {
  return __builtin_amdgcn_wmma_f32_16x16x4_f32(false, a, false, b, (short)0, c,
                                               false, false);
}

__device__ __forceinline__ void atomic_add_f32(float* p, float v) {
  __hip_atomic_fetch_add(p, v, __ATOMIC_RELAXED, __HIP_MEMORY_SCOPE_AGENT);
}

// ---------------------------------------------------------------------------
// Prep: W1T[k][j] = W1[j][k] (k<128), W2T[k][j] = W2[j][k] (k<256)
// ---------------------------------------------------------------------------
__global__ __launch_bounds__(256) void prep_transpose(
    const float* __restrict__ W1, const float* __restrict__ W2,
    float* __restrict__ w1t, float* __restrict__ w2t) {
  int i = blockIdx.x * 256 + threadIdx.x;
  if (i < 128 * 128) {
    int k = i >> 7, j = i & 127;
    w1t[i] = W1[j * K1 + k];
  }
  int i2 = i - 128 * 128;
  if (i2 >= 0 && i2 < 256 * 128) {
    int k = i2 >> 7, j = i2 & 127;
    w2t[i2] = W2[j * K2 + k];
  }
}

// ---------------------------------------------------------------------------
// Edge kernel (persistent, double-buffered): per tile of 16 edges
//   tmp = leaky(concat(h[src], ew) @ W1.T + b1); atomic scatter into sums.
// 128 threads = 4 waves; wave w owns cols [32w, 32w+32).
// All 64 B fragments live in registers for the whole kernel.
// ---------------------------------------------------------------------------
__global__ __launch_bounds__(128) void edge_gemm_scatter(
    const float* __restrict__ h, const int* __restrict__ esrc,
    const int* __restrict__ edst, const float* __restrict__ ew,
    const float* __restrict__ W1, const float* __restrict__ bias1,
    const float* __restrict__ w1t, float* __restrict__ sums,
    float* __restrict__ counts) {
  __shared__ __align__(16) float m_tile[2][16 * 132];  // stride 132: bank-clean
  __shared__ float ew_t[2][16 * 3];
  __shared__ int dst_t[2][16];

  const int tid = threadIdx.x;
  const int wave = tid >> 5, lane = tid & 31;
  const int hh = lane >> 4, l16 = lane & 15;
  const int n0 = wave * 32, n1 = n0 + 16;

  // ---- preload all B fragments into registers (shared by every tile) ----
  v2f B0[32], B1[32];
#pragma unroll
  for (int s = 0; s < 32; ++s) {
    const int kb = 4 * s + 2 * hh;
    B0[s].x = w1t[kb * 128 + n0 + l16];
    B0[s].y = w1t[(kb + 1) * 128 + n0 + l16];
    B1[s].x = w1t[kb * 128 + n1 + l16];
    B1[s].y = w1t[(kb + 1) * 128 + n1 + l16];
  }
  // epilogue constants: edge_w columns of W1 and bias
  const float wc00 = W1[(n0 + l16) * K1 + 128];
  const float wc01 = W1[(n0 + l16) * K1 + 129];
  const float wc02 = W1[(n0 + l16) * K1 + 130];
  const float wc10 = W1[(n1 + l16) * K1 + 128];
  const float wc11 = W1[(n1 + l16) * K1 + 129];
  const float wc12 = W1[(n1 + l16) * K1 + 130];
  const float bb0 = bias1[n0 + l16];
  const float bb1 = bias1[n1 + l16];

  const int e = tid >> 3, q = tid & 7;  // 8 threads gather one h row

  int t = (int)blockIdx.x;
  int it = 0;
  int nx_dst = 0;
  float nx_e0 = 0.f, nx_e1 = 0.f, nx_e2 = 0.f;
#if !HAVE_ASYNC_LDS
  float4 rh0, rh1, rh2, rh3;
#endif

  // ---- prologue: begin staging first tile into buffer 0 ----
  if (t < EDGE_TILES) {
    const int src = esrc[t * 16 + e];
    const float* hr = h + (size_t)src * IN_FEAT;
#if HAVE_ASYNC_LDS
    float* dr = &m_tile[0][e * 132];
    async_b128(hr + q * 4, dr + q * 4);
    async_b128(hr + (q + 8) * 4, dr + (q + 8) * 4);
    async_b128(hr + (q + 16) * 4, dr + (q + 16) * 4);
    async_b128(hr + (q + 24) * 4, dr + (q + 24) * 4);
#else
    const float4* hr4 = (const float4*)hr;
    rh0 = hr4[q]; rh1 = hr4[q + 8]; rh2 = hr4[q + 16]; rh3 = hr4[q + 24];
#endif
    if (tid < 16) {
      nx_dst = edst[t * 16 + tid];
      nx_e0 = ew[(t * 16 + tid) * 3 + 0];
      nx_e1 = ew[(t * 16 + tid) * 3 + 1];
      nx_e2 = ew[(t * 16 + tid) * 3 + 2];
    }
  }

  for (; t < EDGE_TILES; t += (int)gridDim.x, ++it) {
    const int cur = it & 1;
    const float* mt = m_tile[cur];

    // commit metadata for the current tile
    if (tid < 16) {
      dst_t[cur][tid] = nx_dst;
      ew_t[cur][tid * 3 + 0] = nx_e0;
      ew_t[cur][tid * 3 + 1] = nx_e1;
      ew_t[cur][tid * 3 + 2] = nx_e2;
    }
#if HAVE_ASYNC_LDS
    wait_async0();  // h tile for buffer `cur` fully landed in LDS
#else
    float4* dr4 = (float4*)(m_tile[cur] + e * 132);
    dr4[q] = rh0; dr4[q + 8] = rh1; dr4[q + 16] = rh2; dr4[q + 24] = rh3;
#endif
    __syncthreads();

    // ---- begin staging the NEXT tile into the other buffer ----
    const int tn = t + (int)gridDim.x;
    if (tn < EDGE_TILES) {
      const int src = esrc[tn * 16 + e];
      const float* hr = h + (size_t)src * IN_FEAT;
#if HAVE_ASYNC_LDS
      float* dr = &m_tile[cur ^ 1][e * 132];
      async_b128(hr + q * 4, dr + q * 4);
      async_b128(hr + (q + 8) * 4, dr + (q + 8) * 4);
      async_b128(hr + (q + 16) * 4, dr + (q + 16) * 4);
      async_b128(hr + (q + 24) * 4, dr + (q + 24) * 4);
#else
      const float4* hr4 = (const float4*)hr;
      rh0 = hr4[q]; rh1 = hr4[q + 8]; rh2 = hr4[q + 16]; rh3 = hr4[q + 24];
#endif
      if (tid < 16) {
        nx_dst = edst[tn * 16 + tid];
        nx_e0 = ew[(tn * 16 + tid) * 3 + 0];
        nx_e1 = ew[(tn * 16 + tid) * 3 + 1];
        nx_e2 = ew[(tn * 16 + tid) * 3 + 2];
      }
    }

    // ---- 64 WMMAs from LDS A-frags + register B-frags ----
    v8f c0 = {}; v8f c1 = {};
#pragma unroll
    for (int s = 0; s < 32; ++s) {
      const int kb = 4 * s + 2 * hh;
      v2f a;
      {
        const float2 u = *(const float2*)(mt + l16 * 132 + kb);
        a.x = u.x; a.y = u.y;
      }
      c0 = wmma4(a, B0[s], c0);
      c1 = wmma4(a, B1[s], c1);
    }

    // ---- epilogue: edge_w rank-3 term + bias + leakyReLU + scatter ----
#pragma unroll
    for (int r = 0; r < 8; ++r) {
      const int M = r + 8 * hh;
      const int d = dst_t[cur][M];
      const float w0 = ew_t[cur][M * 3 + 0];
      const float w1e = ew_t[cur][M * 3 + 1];
      const float w2e = ew_t[cur][M * 3 + 2];
      float v0 = c0[r] + w0 * wc00 + w1e * wc01 + w2e * wc02 + bb0;
      float v1 = c1[r] + w0 * wc10 + w1e * wc11 + w2e * wc12 + bb1;
      v0 = (v0 >= 0.f) ? v0 : 0.01f * v0;
      v1 = (v1 >= 0.f) ? v1 : 0.01f * v1;
      atomic_add_f32(&sums[(size_t)d * INTER_DIM + n0 + l16], v0);
      atomic_add_f32(&sums[(size_t)d * INTER_DIM + n1 + l16], v1);
    }
    if (tid < 16) atomic_add_f32(&counts[dst_t[cur][tid]], 1.0f);
  }
}

// ---------------------------------------------------------------------------
// Node kernel: out = relu(concat(h, sums/max(cnt,1)) @ W2.T + b2)
// 32 nodes per block (2 M-tiles) so each B-frag load feeds 2 WMMAs.
// ---------------------------------------------------------------------------
__global__ __launch_bounds__(128) void node_gemm(
    const float* __restrict__ h, const float* __restrict__ sums,
    const float* __restrict__ counts, const float* __restrict__ w2t,
    const float* __restrict__ bias2, float* __restrict__ out) {
  __shared__ __align__(16) float a_tile[32 * 260];  // stride 260: bank-clean

  const int node0 = blockIdx.x * 32;
  const int tid = threadIdx.x;

  // ---- stage concat(h, h_N) for 32 nodes (clamped for tail block) ----
  {
    const int nd = tid >> 2, q = tid & 3;  // 4 threads per row
    int node = node0 + nd;
    node = (node < N_NODES) ? node : (N_NODES - 1);
    const float4* hrow = (const float4*)(h + (size_t)node * IN_FEAT);
    float4* drow = (float4*)(a_tile + nd * 260);
#pragma unroll
    for (int i = 0; i < 8; ++i) drow[q + 4 * i] = hrow[q + 4 * i];
    const float inv = 1.0f / fmaxf(counts[node], 1.0f);
    const float4* srow = (const float4*)(sums + (size_t)node * INTER_DIM);
    float4* drow2 = (float4*)(a_tile + nd * 260 + 128);
#pragma unroll
    for (int i = 0; i < 8; ++i) {
      float4 v = srow[q + 4 * i];
      v.x *= inv; v.y *= inv; v.z *= inv; v.w *= inv;
      drow2[q + 4 * i] = v;
    }
  }
  __syncthreads();

  const int wave = tid >> 5, lane = tid & 31;
  const int hh = lane >> 4, l16 = lane & 15;
  const int n0 = wave * 32, n1 = n0 + 16;

  v8f c00 = {}, c01 = {}, c10 = {}, c11 = {};
#pragma unroll
  for (int s = 0; s < 64; ++s) {
    const int kb = 4 * s + 2 * hh;
    v2f a0, a1;
    {
      const float2 u = *(const float2*)(a_tile + l16 * 260 + kb);
      a0.x = u.x; a0.y = u.y;
    }
    {
      const float2 u = *(const float2*)(a_tile + (16 + l16) * 260 + kb);
      a1.x = u.x; a1.y = u.y;
    }
    v2f b0, b1f;
    b0.x = w2t[kb * 128 + n0 + l16];
    b0.y = w2t[(kb + 1) * 128 + n0 + l16];
    b1f.x = w2t[kb * 128 + n1 + l16];
    b1f.y = w2t[(kb + 1) * 128 + n1 + l16];
    c00 = wmma4(a0, b0, c00);
    c01 = wmma4(a0, b1f, c01);
    c10 = wmma4(a1, b0, c10);
    c11 = wmma4(a1, b1f, c11);
  }

  const float bb0 = bias2[n0 + l16];
  const float bb1 = bias2[n1 + l16];
#pragma unroll
  for (int r = 0; r < 8; ++r) {
    const int M = r + 8 * hh;
    const int nA = node0 + M;
    const int nB = node0 + 16 + M;
    if (nA < N_NODES) {
      out[(size_t)nA * OUT_FEAT + n0 + l16] = fmaxf(c00[r] + bb0, 0.f);
      out[(size_t)nA * OUT_FEAT + n1 + l16] = fmaxf(c01[r] + bb1, 0.f);
    }
    if (nB < N_NODES) {
      out[(size_t)nB * OUT_FEAT + n0 + l16] = fmaxf(c10[r] + bb0, 0.f);
      out[(size_t)nB * OUT_FEAT + n1 + l16] = fmaxf(c11[r] + bb1, 0.f);
    }
  }
}

// ---------------------------------------------------------------------------
extern "C" void kernel_launch(void* const* d_in, const int* in_sizes, int n_in,
                              void* d_out, int out_size, void* d_ws, size_t ws_size,
                              hipStream_t stream) {
  const float* h    = (const float*)d_in[0];
  const int*   esrc = (const int*)d_in[1];
  const int*   edst = (const int*)d_in[2];
  const float* ew   = (const float*)d_in[3];
  const float* W1   = (const float*)d_in[4];
  const float* b1   = (const float*)d_in[5];
  const float* W2   = (const float*)d_in[6];
  const float* b2   = (const float*)d_in[7];
  float* out = (float*)d_out;

  // workspace layout
  float* sums   = (float*)d_ws;                          // 50000*128
  float* counts = sums + (size_t)N_NODES * INTER_DIM;    // 50000
  float* w1t    = counts + N_NODES;                      // 128*128
  float* w2t    = w1t + 128 * 128;                       // 256*128

  (void)hipMemsetAsync(sums, 0,
                       ((size_t)N_NODES * INTER_DIM + N_NODES) * sizeof(float),
                       stream);
  prep_transpose<<<(128 * 128 + 256 * 128 + 255) / 256, 256, 0, stream>>>(
      W1, W2, w1t, w2t);
  edge_gemm_scatter<<<EDGE_BLOCKS, 128, 0, stream>>>(
      h, esrc, edst, ew, W1, b1, w1t, sums, counts);
  node_gemm<<<(N_NODES + 31) / 32, 128, 0, stream>>>(
      h, sums, counts, w2t, b2, out);
}